// InstanceColouringSBP_36730560315754
// MI455X (gfx1250) — compile-verified
//
#include <hip/hip_runtime.h>

#define B_    32
#define FEAT_ 512
#define S_    128
#define S2_   (S_ * S_)      // 16384
#define CD_   8
#define K_    7

typedef __attribute__((ext_vector_type(2))) float v2f;
typedef __attribute__((ext_vector_type(8))) float v8f;

// ---------------------------------------------------------------------------
// Kernel 1: x = gate*(W @ features + b); colour = x + uv; delta = x[:, 6:8]
// GEMM per batch: [16(pad of 8) x 512] @ [512 x 16384] via V_WMMA_F32_16X16X4_F32
// One wave = one 16-pixel tile. 8 waves/block = 128-pixel slab.
// ---------------------------------------------------------------------------
__global__ __launch_bounds__(256) void conv_colour_kernel(
    const float* __restrict__ features,   // (B, FEAT, S, S)
    const float* __restrict__ conv_w,     // (CD, FEAT)
    const float* __restrict__ conv_b,     // (CD,)
    const float* __restrict__ gate,       // (1,)
    const float* __restrict__ uv,         // (1, CD, S, S)
    float* __restrict__ colour,           // (B, CD, S, S)
    float* __restrict__ delta)            // (B, 2, S, S)
{
    __shared__ float w_lds[16 * FEAT_];   // rows 8..15 zero-padded (32 KB)

    const int tid = threadIdx.x;
    for (int i = tid; i < CD_ * FEAT_; i += 256) w_lds[i] = conv_w[i];
    for (int i = tid; i < CD_ * FEAT_; i += 256) w_lds[CD_ * FEAT_ + i] = 0.0f;
    __syncthreads();

    const int b    = blockIdx.x >> 7;     // 128 slabs per batch
    const int slab = blockIdx.x & 127;
    const int wave = tid >> 5;
    const int lane = tid & 31;
    const int m    = lane & 15;           // A: M index (channel); B/D: N index (pixel)
    const int kh   = lane >> 4;           // K half-select per ISA 16x4 layout
    const int pix  = slab * 128 + wave * 16 + m;

    const float* fbase = features + (size_t)b * FEAT_ * S2_ + pix;

    v8f acc = {};
    #pragma unroll 4
    for (int kb = 0; kb < FEAT_; kb += 4) {
        // A fragment: lane m holds W[m][kb + 2*kh + {0,1}]  (ds_load_b64)
        const v2f a = *(const v2f*)&w_lds[m * FEAT_ + kb + 2 * kh];
        // B fragment: lane holds F[b][kb + 2*kh + {0,1}][pix] (coalesced b32 x2)
        v2f bb;
        bb.x = fbase[(size_t)(kb + 2 * kh)     * S2_];
        bb.y = fbase[(size_t)(kb + 2 * kh + 1) * S2_];
        acc = __builtin_amdgcn_wmma_f32_16x16x4_f32(
            /*neg_a=*/false, a, /*neg_b=*/false, bb,
            /*c_mod=*/(short)0, acc, /*reuse_a=*/false, /*reuse_b=*/false);
    }

    // Lanes 0..15: acc[j] = channel j at pixel (tile_base + lane)
    if (lane < 16) {
        const float g = gate[0];
        #pragma unroll
        for (int ch = 0; ch < CD_; ++ch) {
            const float xv = g * (acc[ch] + conv_b[ch]);
            const float cv = xv + uv[ch * S2_ + pix];
            colour[((size_t)b * CD_ + ch) * S2_ + pix] = cv;
            if (ch >= CD_ - 2)
                delta[((size_t)b * 2 + (ch - (CD_ - 2))) * S2_ + pix] = xv;
        }
    }
}

// ---------------------------------------------------------------------------
// Kernel 2: per-batch argmax of rand_pixel * exp(log_s); gather seed colour.
// One block per batch.
// ---------------------------------------------------------------------------
__global__ __launch_bounds__(256) void argmax_seed_kernel(
    const float* __restrict__ rand_pixel, // (B, 1, S, S)
    const float* __restrict__ log_s,      // slice k: (B, 1, S, S)
    const float* __restrict__ colour,     // (B, CD, S, S)
    float* __restrict__ seeds)            // slice k: (B, CD)
{
    __shared__ float sval[256];
    __shared__ int   sidx[256];

    const int b   = blockIdx.x;
    const int tid = threadIdx.x;
    const float* rp = rand_pixel + (size_t)b * S2_;
    const float* ls = log_s      + (size_t)b * S2_;

    float best = -1.0f;
    int   bidx = 0;
    for (int p = tid; p < S2_; p += 256) {
        const float v = rp[p] * expf(ls[p]);
        if (v > best) { best = v; bidx = p; }
    }
    sval[tid] = best;
    sidx[tid] = bidx;
    __syncthreads();

    for (int s = 128; s > 0; s >>= 1) {
        if (tid < s) {
            const float v2 = sval[tid + s];
            const int   i2 = sidx[tid + s];
            if (v2 > sval[tid] || (v2 == sval[tid] && i2 < sidx[tid])) {
                sval[tid] = v2; sidx[tid] = i2;
            }
        }
        __syncthreads();
    }

    if (tid < CD_) {
        const int idx = sidx[0];
        seeds[b * CD_ + tid] = colour[((size_t)b * CD_ + tid) * S2_ + idx];
    }
}

// ---------------------------------------------------------------------------
// Kernel 3: elementwise scan step.
//   d2 = sum_c (colour - seed)^2 ; alpha = clamp(exp(-d2/sigma), .01, .99)
//   log_m[k] = log_s + log(alpha) ; log_s[k+1] = log_s + log1p(-alpha)
// 64 blocks per batch -> seed loads are block-uniform (scalar path).
// ---------------------------------------------------------------------------
__global__ __launch_bounds__(256) void update_kernel(
    const float* __restrict__ colour,     // (B, CD, S, S)
    const float* __restrict__ seeds,      // slice k: (B, CD)
    const float* __restrict__ log_s_in,   // log_s_k slice k
    float* __restrict__ log_m_out,        // log_m_k slice k
    float* __restrict__ log_s_out,        // log_s_k slice k+1
    float* __restrict__ log_m_final,      // log_m_k slice K on last step, else null
    const float* __restrict__ log_sigma)  // (1,)
{
    const int b = blockIdx.x >> 6;
    const int p = ((blockIdx.x & 63) << 8) + threadIdx.x;

    const float inv_sigma = expf(-log_sigma[0]);
    const float* cb = colour + (size_t)b * CD_ * S2_;

    float d2 = 0.0f;
    #pragma unroll
    for (int ch = 0; ch < CD_; ++ch) {
        const float d = cb[(size_t)ch * S2_ + p] - seeds[b * CD_ + ch];
        d2 += d * d;
    }

    float alpha = expf(-d2 * inv_sigma);
    alpha = fminf(fmaxf(alpha, 0.01f), 0.99f);

    const size_t off = (size_t)b * S2_ + p;
    const float ls  = log_s_in[off];
    const float nls = ls + log1pf(-alpha);
    log_m_out[off] = ls + logf(alpha);
    log_s_out[off] = nls;
    if (log_m_final) log_m_final[off] = nls;   // log_m_k[K] = final log_s
}

__global__ __launch_bounds__(256) void zero_kernel(float* __restrict__ p, int n)
{
    const int i = blockIdx.x * 256 + threadIdx.x;
    if (i < n) p[i] = 0.0f;
}

// ---------------------------------------------------------------------------
// Output layout (flat, return order):
//   log_m_k (8,32,1,128,128) | log_s_k (8,32,1,128,128) | seeds (7,32,8)
//   | colour (32,8,128,128)  | delta (32,2,128,128)
// ---------------------------------------------------------------------------
extern "C" void kernel_launch(void* const* d_in, const int* in_sizes, int n_in,
                              void* d_out, int out_size, void* d_ws, size_t ws_size,
                              hipStream_t stream) {
    const float* features   = (const float*)d_in[0];
    const float* rand_pixel = (const float*)d_in[1];
    const float* conv_w     = (const float*)d_in[2];
    const float* conv_b     = (const float*)d_in[3];
    const float* gate       = (const float*)d_in[4];
    const float* log_sigma  = (const float*)d_in[5];
    const float* uv         = (const float*)d_in[6];
    // d_in[7] = steps_to_run (== K_, fixed by the reference)

    float* out     = (float*)d_out;
    float* log_m_k = out;                                   // 8*B*S2
    float* log_s_k = out + (size_t)(K_ + 1) * B_ * S2_;     // 8*B*S2
    float* seeds   = log_s_k + (size_t)(K_ + 1) * B_ * S2_; // 7*B*CD
    float* colour  = seeds + (size_t)K_ * B_ * CD_;         // B*CD*S2
    float* delta   = colour + (size_t)B_ * CD_ * S2_;       // B*2*S2

    // log_s_k[0] = 0
    zero_kernel<<<(B_ * S2_ + 255) / 256, 256, 0, stream>>>(log_s_k, B_ * S2_);

    // conv + colour + delta (WMMA)
    conv_colour_kernel<<<B_ * 128, 256, 0, stream>>>(
        features, conv_w, conv_b, gate, uv, colour, delta);

    // sequential scan: K steps of (argmax/seed, elementwise update)
    for (int k = 0; k < K_; ++k) {
        argmax_seed_kernel<<<B_, 256, 0, stream>>>(
            rand_pixel, log_s_k + (size_t)k * B_ * S2_, colour,
            seeds + (size_t)k * B_ * CD_);
        update_kernel<<<B_ * 64, 256, 0, stream>>>(
            colour, seeds + (size_t)k * B_ * CD_,
            log_s_k + (size_t)k * B_ * S2_,
            log_m_k + (size_t)k * B_ * S2_,
            log_s_k + (size_t)(k + 1) * B_ * S2_,
            (k == K_ - 1) ? (log_m_k + (size_t)K_ * B_ * S2_) : nullptr,
            log_sigma);
    }
}